// BioSelfAttention_41257455845997
// MI455X (gfx1250) — compile-verified
//
#include <hip/hip_runtime.h>
#include <math.h>

typedef __attribute__((ext_vector_type(2))) float v2f;
typedef __attribute__((ext_vector_type(8))) float v8f;

#define EPS_F   1e-9f
#define T_DIM   4096
#define D_DIM   64
#define BH      64          // B*H = 4*16

// -------------------------------------------------------------------------
// Kernel 1: zero the whole 64 MiB output with 128-bit stores (HBM-store bound)
// -------------------------------------------------------------------------
__global__ void bio_zero_fill_f4(float4* __restrict__ out, size_t n4) {
    size_t i      = (size_t)blockIdx.x * blockDim.x + threadIdx.x;
    size_t stride = (size_t)gridDim.x * blockDim.x;
    float4 z; z.x = 0.f; z.y = 0.f; z.z = 0.f; z.w = 0.f;
    for (; i < n4; i += stride) out[i] = z;
}

// -------------------------------------------------------------------------
// Kernel 2: one workgroup (4 waves). Wave w owns (b,h) rows 16w..16w+15.
// WMMA f32 16x16x4 accumulates QK^T, QQ^T, KK^T over D=64 in 16 chunks of 4;
// diagonals give Q.K, |Q|^2, |K|^2 per row. Then the iterated map (exact
// fixed-point early exit == full 4096-step scan in f32), then scale V[t=0].
// -------------------------------------------------------------------------
__global__ __launch_bounds__(128)
void bio_attn_t0(const float* __restrict__ Q,
                 const float* __restrict__ K,
                 const float* __restrict__ V,
                 float* __restrict__ out) {
    __shared__ float s_qk[BH];
    __shared__ float s_qq[BH];
    __shared__ float s_kk[BH];
    __shared__ float s_r [BH];

    const int tid  = threadIdx.x;
    const int wave = tid >> 5;
    const int lane = tid & 31;
    const int m16  = lane & 15;
    const size_t rowStride = (size_t)T_DIM * D_DIM;          // floats per (b,h)
    const size_t rowBase   = (size_t)(wave * 16 + m16) * rowStride;
    const int kbase = (lane < 16) ? 0 : 2;                   // A/B k-half split

    v8f acc_qk = {}; v8f acc_qq = {}; v8f acc_kk = {};
#pragma unroll
    for (int c = 0; c < 16; ++c) {
        const size_t off = rowBase + (size_t)(4 * c + kbase);
        v2f q = *(const v2f*)(Q + off);   // serves as A tile AND B(=Q^T) tile
        v2f k = *(const v2f*)(K + off);
        acc_qk = __builtin_amdgcn_wmma_f32_16x16x4_f32(false, q, false, k,
                                                       (short)0, acc_qk, false, false);
        acc_qq = __builtin_amdgcn_wmma_f32_16x16x4_f32(false, q, false, q,
                                                       (short)0, acc_qq, false, false);
        acc_kk = __builtin_amdgcn_wmma_f32_16x16x4_f32(false, k, false, k,
                                                       (short)0, acc_kk, false, false);
    }

    // C/D layout: vgpr j, lanes 0-15 -> (M=j, N=lane); lanes 16-31 -> (M=j+8, N=lane-16).
    // Diagonal (M==N): lanes 0..7 hold it in vgpr=lane; lanes 24..31 in vgpr=lane-24.
    const bool hasDiag = (lane < 8) || (lane >= 24);
    const int  jsel    = (lane < 8) ? lane : (lane - 24);
    const int  mrow    = (lane < 8) ? lane : (lane - 16);
    float dqk = 0.f, dqq = 0.f, dkk = 0.f;
#pragma unroll
    for (int j = 0; j < 8; ++j) {
        if (j == jsel) { dqk = acc_qk[j]; dqq = acc_qq[j]; dkk = acc_kk[j]; }
    }
    if (hasDiag) {
        const int idx = wave * 16 + mrow;
        s_qk[idx] = dqk; s_qq[idx] = dqq; s_kk[idx] = dkk;
    }
    __syncthreads();

    if (lane < 16) {
        const int idx = wave * 16 + lane;
        const float qk = s_qk[idx];
        const float qq = s_qq[idx];
        const float kk = s_kk[idx];
        float r = qk / ((sqrtf(qq) + EPS_F) * (sqrtf(kk) + EPS_F));
        // scan: r <- r/(r+eps), 4096 times; f(x)==x is an exact fixed point in
        // f32, so early exit is bit-identical to running all iterations.
#pragma unroll 1
        for (int i = 0; i < T_DIM; ++i) {
            const float n = r / (r + EPS_F);
            if (n == r) break;
            r = n;
        }
        s_r[idx] = r;
    }
    __syncthreads();

    // Write the 64 scaled V rows: thread t -> row t>>1, 32-float half t&1.
    {
        const int   row  = tid >> 1;
        const int   half = tid & 1;
        const float r    = s_r[row];
        const size_t base = (size_t)row * rowStride + (size_t)half * 32;
        const float4* vp = (const float4*)(V + base);
        float4*       op = (float4*)(out + base);
#pragma unroll
        for (int i = 0; i < 8; ++i) {
            float4 v = vp[i];
            float4 o;
            o.x = r * v.x; o.y = r * v.y; o.z = r * v.z; o.w = r * v.w;
            op[i] = o;
        }
    }
}

// -------------------------------------------------------------------------
extern "C" void kernel_launch(void* const* d_in, const int* in_sizes, int n_in,
                              void* d_out, int out_size, void* d_ws, size_t ws_size,
                              hipStream_t stream) {
    const float* Q = (const float*)d_in[0];
    const float* K = (const float*)d_in[1];
    const float* V = (const float*)d_in[2];
    float* out = (float*)d_out;

    // Zero fill (covers everything, incl. t=0 rows which kernel 2 overwrites
    // afterwards in stream order).
    const size_t n4 = (size_t)out_size / 4;                 // float4 count
    const int threads = 256;
    int blocks = (int)((n4 + (size_t)threads * 4 - 1) / ((size_t)threads * 4));
    if (blocks < 1) blocks = 1;
    bio_zero_fill_f4<<<blocks, threads, 0, stream>>>((float4*)out, n4);

    // Tiny compute kernel: 1 workgroup, 4 waves (one per 16 (b,h) rows).
    bio_attn_t0<<<1, 128, 0, stream>>>(Q, K, V, out);
}